// CudnnWaterBalanceNN_36094905155769
// MI455X (gfx1250) — compile-verified
//
#include <hip/hip_runtime.h>

typedef float v2f __attribute__((ext_vector_type(2)));
typedef float v8f __attribute__((ext_vector_type(8)));

#define ITER 512
#define NBATCH 65536
#define DT 0.1f

// Per wave: 32 batch elements as two 16-batch WMMA pipelines (A, B).
// Layer 1 (transposed):  H^T tile = A1_t (16 hid x 4) * Bx (4 x 16 batch), K slots:
//   A1 vgpr0: lanes0-15 (K0) = W1[0, 16t+m], lanes16-31 (K2) = b1[16t+m]
//   A1 vgpr1: lanes0-15 (K1) = W1[1, 16t+m], lanes16-31 (K3) = 0
//   Bx vgpr0: lanes0-15 (K0) = inflow[n],    lanes16-31 (K2) = 1.0   (bias row)
//   Bx vgpr1: lanes0-15 (K1) = storage[n],   lanes16-31 (K3) = 0
// Layer 2: the C-layout of H^T doubles as a B-layout. VGPR pair (h[2p],h[2p+1])
// is a 4x16 B with K-rows = hidden {2p,2p+1,2p+8,2p+9} of tile t, N = batch.
// A2_{t,p} holds matching w2[j_k] (uniform over M), so the chained WMMAs
// accumulate outflow[batch] replicated across all M rows / both lane halves.

__device__ __forceinline__ float relu1(float x) {
  float y;
  // single exact relu; avoids clang's IEEE canonicalize (v_max x,x,x) prefix
  asm("v_max_num_f32 %0, 0, %1" : "=v"(y) : "v"(x));
  return y;
}

__global__ __launch_bounds__(256) void water_balance_wmma_kernel(
    const float* __restrict__ inflows, const float* __restrict__ storage0,
    const float* __restrict__ W1, const float* __restrict__ b1,
    const float* __restrict__ W2, const float* __restrict__ b2,
    float* __restrict__ storages_out, float* __restrict__ outflows_out)
{
  const int lane = threadIdx.x & 31;
  const int wave = (int)((blockIdx.x * blockDim.x + threadIdx.x) >> 5);
  const int b32  = wave * 32;
  const int lm   = lane & 15;
  const bool lo  = (lane < 16);
  const int half8 = lo ? 0 : 8;

  // ---- preload layer-1 A tiles (32 VGPRs) ----
  v2f a1[16];
#pragma unroll
  for (int t = 0; t < 16; ++t) {
    const int j = 16 * t + lm;
    const float* p0 = lo ? (W1 + j) : (b1 + j);   // K0 | K2(bias)
    float x0 = *p0;
    float x1 = lo ? W1[256 + j] : 0.0f;           // K1 | K3(=0)
    a1[t].x = x0;
    a1[t].y = x1;
  }

  // ---- preload layer-2 w2-selector A tiles (128 VGPRs, 64 b64 loads) ----
  v2f a2[64];
#pragma unroll
  for (int t = 0; t < 16; ++t) {
#pragma unroll
    for (int p = 0; p < 4; ++p) {
      const int j = 16 * t + 2 * p + half8;       // even -> 8B aligned pair
      a2[t * 4 + p] = *(const v2f*)(W2 + j);      // {K0|K2, K1|K3}
    }
  }

  const float sb2 = b2[0];
  float sA = storage0[b32 + lm];        // batch lm,    replicated across lanes
  float sB = storage0[b32 + 16 + lm];   // batch 16+lm, replicated across lanes

  // storages row 0 = storage0
  __builtin_nontemporal_store(storage0[b32 + lane], &storages_out[b32 + lane]);

  const v8f z = {0.f, 0.f, 0.f, 0.f, 0.f, 0.f, 0.f, 0.f};

  // software-pipelined inflow stream (streamed once -> non-temporal)
  float inext = __builtin_nontemporal_load(&inflows[b32 + lane]);

#pragma unroll 1
  for (int t = 0; t < ITER; ++t) {
    const float icur = inext;
    const int tn = (t + 1 < ITER) ? (t + 1) : (ITER - 1);
    inext = __builtin_nontemporal_load(&inflows[(size_t)tn * NBATCH + b32 + lane]);
    const int tp = (t + 8 < ITER) ? (t + 8) : (ITER - 1);
    __builtin_prefetch(&inflows[(size_t)tp * NBATCH + b32 + lane], 0, 3);

    // swap lane halves so both pipelines see their 16 batch values everywhere
    const float ihi =
        __builtin_bit_cast(float, __builtin_amdgcn_ds_swizzle(
                                      __builtin_bit_cast(int, icur), 0x401F)); // SWAPX16
    const float inflA = lo ? icur : ihi;   // batch lm everywhere
    const float inflB = lo ? ihi : icur;   // batch 16+lm everywhere

    v2f bA, bB;
    bA.x = lo ? inflA : 1.0f;  bA.y = lo ? sA : 0.0f;
    bB.x = lo ? inflB : 1.0f;  bB.y = lo ? sB : 0.0f;

    v8f accA[4], accB[4];
#pragma unroll
    for (int q = 0; q < 16; ++q) {
      // ---- pipeline A: layer1 tile, relu, 4 layer2 accumulating WMMAs ----
      v8f h = __builtin_amdgcn_wmma_f32_16x16x4_f32(false, a1[q], false, bA,
                                                    (short)0, z, false, false);
#pragma unroll
      for (int e = 0; e < 8; ++e) h[e] = relu1(h[e]);
#pragma unroll
      for (int p = 0; p < 4; ++p) {
        v2f hb; hb.x = h[2 * p]; hb.y = h[2 * p + 1];
        accA[p] = __builtin_amdgcn_wmma_f32_16x16x4_f32(
            false, a2[q * 4 + p], false, hb, (short)0,
            (q == 0) ? z : accA[p], false, false);
      }
      // ---- pipeline B ----
      v8f g = __builtin_amdgcn_wmma_f32_16x16x4_f32(false, a1[q], false, bB,
                                                    (short)0, z, false, false);
#pragma unroll
      for (int e = 0; e < 8; ++e) g[e] = relu1(g[e]);
#pragma unroll
      for (int p = 0; p < 4; ++p) {
        v2f gb; gb.x = g[2 * p]; gb.y = g[2 * p + 1];
        accB[p] = __builtin_amdgcn_wmma_f32_16x16x4_f32(
            false, a2[q * 4 + p], false, gb, (short)0,
            (q == 0) ? z : accB[p], false, false);
      }
    }

    // D is replicated over M and lane halves: element 0 of any acc VGPR works
    const float outA = ((accA[0][0] + accA[1][0]) + (accA[2][0] + accA[3][0])) + sb2;
    const float outB = ((accB[0][0] + accB[1][0]) + (accB[2][0] + accB[3][0])) + sb2;

    sA = __builtin_fmaf(DT, inflA - outA, sA);
    sB = __builtin_fmaf(DT, inflB - outB, sB);

    __builtin_nontemporal_store(lo ? outA : outB,
                                &outflows_out[(size_t)t * NBATCH + b32 + lane]);
    __builtin_nontemporal_store(lo ? sA : sB,
                                &storages_out[(size_t)(t + 1) * NBATCH + b32 + lane]);
  }
}

extern "C" void kernel_launch(void* const* d_in, const int* in_sizes, int n_in,
                              void* d_out, int out_size, void* d_ws, size_t ws_size,
                              hipStream_t stream) {
  const float* inflows  = (const float*)d_in[0];
  const float* storage0 = (const float*)d_in[1];
  const float* W1       = (const float*)d_in[2];
  const float* b1       = (const float*)d_in[3];
  const float* W2       = (const float*)d_in[4];
  const float* b2       = (const float*)d_in[5];

  float* storages = (float*)d_out;                               // (513, 65536)
  float* outflows = (float*)d_out + (size_t)(ITER + 1) * NBATCH; // (512, 65536)

  // 65536 batch / 32 per wave = 2048 waves; 8 waves (256 thr) per block
  dim3 grid(NBATCH / 32 / 8);
  dim3 block(256);
  water_balance_wmma_kernel<<<grid, block, 0, stream>>>(
      inflows, storage0, W1, b1, W2, b2, storages, outflows);
}